// BoundaryLoss_11184094838828
// MI455X (gfx1250) — compile-verified
//
#include <hip/hip_runtime.h>
#include <math.h>

// Problem constants (match reference)
#define BB   2
#define CC   3
#define NN   (96 * 96 * 96)      // 884736 voxels per volume
#define BIGF 1.0e8f
#define NBLK 864                 // accumulation blocks (NN / 1024)

typedef __attribute__((ext_vector_type(2))) float v2f;
typedef __attribute__((ext_vector_type(8))) float v8f;

// ---------------------------------------------------------------------------
// init: zero per-volume flags and the block-sum accumulator slots
// ---------------------------------------------------------------------------
__global__ void bl_init(int* __restrict__ flags, float* __restrict__ blockSums) {
    int i = blockIdx.x * blockDim.x + threadIdx.x;
    if (i < 4) flags[i] = 0;
    if (i < NBLK) blockSums[i] = 0.0f;
}

// ---------------------------------------------------------------------------
// mask: build the two EDT source fields for volume v = (b, cls)
//   F[0..N)  = pos ? 0    : BIG   (source for edt(~pos) -> dist_out)
//   F[N..2N) = pos ? BIG  : 0     (source for edt(pos)  -> dist_in)
// and record pos.any() per volume (one atomicOr per block).
// ---------------------------------------------------------------------------
__global__ void bl_mask(float* __restrict__ F, const int* __restrict__ tg,
                        int* __restrict__ flags, int b, int cls, int v) {
    __shared__ int bf;
    if (threadIdx.x == 0) bf = 0;
    __syncthreads();
    int x = blockIdx.x * blockDim.x + threadIdx.x;
    bool pos = false;
    if (x < NN) {
        pos = (tg[b * NN + x] == cls);
        F[x]      = pos ? 0.0f : BIGF;
        F[NN + x] = pos ? BIGF : 0.0f;
    }
    if (pos) bf = 1;
    __syncthreads();
    if (threadIdx.x == 0 && bf) atomicOr(flags + v, 1);
}

// ---------------------------------------------------------------------------
// min-conv along W (contiguous lines, stride 1). One 96-float line per
// (ty) slot; line staged in LDS, inner loop is LDS-broadcast reads.
//   g[i] = min_j f[j] + (i-j)^2        (exact match to reference f32 math)
// ---------------------------------------------------------------------------
__global__ void bl_minconv_w(float* __restrict__ F) {
    __shared__ float lds[4][96];
    const int tx = threadIdx.x, ty = threadIdx.y;
    const int base = (blockIdx.x * 4 + ty) * 96;
    lds[ty][tx] = F[base + tx];
    __syncthreads();
    float m = 3.402823e38f;
#pragma unroll 8
    for (int j = 0; j < 96; ++j) {
        float dj = (float)(tx - j);
        m = fminf(m, lds[ty][j] + dj * dj);
    }
    F[base + tx] = m;
}

// ---------------------------------------------------------------------------
// min-conv along a strided axis. Element address: base0 + i*S, where
//   base0 = f*NN + a*Amul + xg*32 + tx
// H pass: Amul = 96*96, S = 96 ; D pass: Amul = 96, S = 96*96.
// 32 consecutive-in-x lines per block -> fully coalesced 128B global
// accesses; 96x32 LDS tile -> bank-conflict-free (lane tx = bank).
// ---------------------------------------------------------------------------
__global__ void bl_minconv_s(float* __restrict__ F, int Amul, int S) {
    __shared__ float tile[96 * 32];
    const int tx = threadIdx.x, ty = threadIdx.y;
    const int xg = blockIdx.x % 3;
    const int a  = (blockIdx.x / 3) % 96;
    const int f  = blockIdx.x / (3 * 96);
    const int base0 = f * NN + a * Amul + xg * 32 + tx;
    for (int i = ty; i < 96; i += 8) tile[i * 32 + tx] = F[base0 + i * S];
    __syncthreads();
    for (int i = ty; i < 96; i += 8) {
        float m = 3.402823e38f;
#pragma unroll 8
        for (int j = 0; j < 96; ++j) {
            float dj = (float)(i - j);
            m = fminf(m, tile[j * 32 + tx] + dj * dj);
        }
        F[base0 + i * S] = m;
    }
}

// ---------------------------------------------------------------------------
// accumulate: softmax prob of class `cls` times signed distance, reduced
// per wave with V_WMMA_F32_16X16X4_F32 (A = lane partials in the 16x4
// layout, B = ones  =>  D[m][n] = acc_m + acc_{m+16}; summing the 8 D
// VGPRs per lane and adding shfl_xor(16) yields the exact wave-32 sum).
// Block sums go to fixed slots (no float atomics -> deterministic).
// ---------------------------------------------------------------------------
__global__ void bl_accum(const float* __restrict__ F, const float* __restrict__ logits,
                         const int* __restrict__ flags, float* __restrict__ blockSums,
                         int b, int cls, int v) {
    const int stride = gridDim.x * blockDim.x;      // 221184 = NN/4: uniform 4 iters
    const float* lb = logits + (size_t)b * CC * NN;
    const float fl = flags[v] ? 1.0f : 0.0f;
    float acc = 0.0f;
    for (int x = blockIdx.x * blockDim.x + threadIdx.x; x < NN; x += stride) {
        float a0 = lb[x], a1 = lb[NN + x], a2 = lb[2 * NN + x];
        float mx = fmaxf(a0, fmaxf(a1, a2));
        float e0 = expf(a0 - mx), e1 = expf(a1 - mx), e2 = expf(a2 - mx);
        float p  = (cls == 1 ? e1 : e2) / (e0 + e1 + e2);
        float sd = (sqrtf(F[x]) - sqrtf(F[NN + x])) * fl;   // dist_out - dist_in
        acc += p * sd;
    }
    // --- wave32 reduction via WMMA f32 16x16x4 (EXEC all-1s here) ---
    v2f va; va[0] = acc;  va[1] = 0.0f;
    v2f vb; vb[0] = 1.0f; vb[1] = 1.0f;
    v8f vc = {0.f, 0.f, 0.f, 0.f, 0.f, 0.f, 0.f, 0.f};
    v8f vd = __builtin_amdgcn_wmma_f32_16x16x4_f32(
        /*neg_a=*/false, va, /*neg_b=*/false, vb,
        /*c_mod=*/(short)0, vc, /*reuse_a=*/false, /*reuse_b=*/false);
    float s8 = vd[0] + vd[1] + vd[2] + vd[3] + vd[4] + vd[5] + vd[6] + vd[7];
    float wtot = s8 + __shfl_xor(s8, 16, 32);       // exact sum of 32 lane partials

    __shared__ float wsum[8];
    const int lane = threadIdx.x & 31, w = threadIdx.x >> 5;
    if (lane == 0) wsum[w] = wtot;
    __syncthreads();
    if (threadIdx.x == 0) {
        float bs = 0.0f;
        for (int i = 0; i < 8; ++i) bs += wsum[i];
        blockSums[blockIdx.x] += bs;                // fixed slot, sequential kernels
    }
}

// ---------------------------------------------------------------------------
// final: deterministic fixed-order sum of the 864 block sums, then the
// mean/B normalization:  out = total / N / B
// ---------------------------------------------------------------------------
__global__ void bl_final(const float* __restrict__ blockSums, float* __restrict__ out) {
    if (blockIdx.x == 0 && threadIdx.x == 0) {
        float t = 0.0f;
        for (int i = 0; i < NBLK; ++i) t += blockSums[i];
        out[0] = t / (float)NN / (float)BB;
    }
}

// ---------------------------------------------------------------------------
extern "C" void kernel_launch(void* const* d_in, const int* in_sizes, int n_in,
                              void* d_out, int out_size, void* d_ws, size_t ws_size,
                              hipStream_t stream) {
    const float* logits  = (const float*)d_in[0];
    const int*   targets = (const int*)d_in[1];
    float* out = (float*)d_out;

    // workspace: 2 EDT fields for one volume (ping volume-at-a-time to keep
    // footprint at ~7.1 MB), then flags[4] (padded to 8), then blockSums[864]
    float* F         = (float*)d_ws;
    int*   flags     = (int*)(F + 2 * NN);
    float* blockSums = (float*)(flags + 8);

    bl_init<<<4, 256, 0, stream>>>(flags, blockSums);

    for (int v = 0; v < BB * (CC - 1); ++v) {
        const int b = v / (CC - 1);
        const int cls = (v % (CC - 1)) + 1;
        bl_mask<<<NN / 256, 256, 0, stream>>>(F, targets, flags, b, cls, v);
        bl_minconv_w<<<(2 * 96 * 96) / 4, dim3(96, 4), 0, stream>>>(F);
        bl_minconv_s<<<576, dim3(32, 8), 0, stream>>>(F, 96 * 96, 96); // H axis
        bl_minconv_s<<<576, dim3(32, 8), 0, stream>>>(F, 96, 96 * 96); // D axis
        bl_accum<<<NBLK, 256, 0, stream>>>(F, logits, flags, blockSums, b, cls, v);
    }
    bl_final<<<1, 64, 0, stream>>>(blockSums, out);
}